// DigitConvolutionalModel_78520592105772
// MI455X (gfx1250) — compile-verified
//
#include <hip/hip_runtime.h>

typedef float v2f __attribute__((ext_vector_type(2)));
typedef float v8f __attribute__((ext_vector_type(8)));

// Tile/layout constants
#define BM   64            // batch rows per workgroup
#define HS   680           // LDS row stride for conv output h [BM][676]
#define H1S  260           // LDS row stride for h1 [BM][256]
#define H2S  132           // LDS row stride for h2 [BM][128]
#define LDS_FLOATS (BM*HS + BM*H1S + BM*H2S)

__global__ void __launch_bounds__(256, 1)
digitnet_fused_kernel(const float* __restrict__ x,      // [B,784]
                      const float* __restrict__ cw,     // [3,3]
                      const float* __restrict__ w1,     // [676,256]
                      const float* __restrict__ b1,     // [256]
                      const float* __restrict__ w2,     // [256,128]
                      const float* __restrict__ b2,     // [128]
                      const float* __restrict__ w3,     // [128,10]
                      const float* __restrict__ b3,     // [10]
                      float* __restrict__ out)          // [B,10]
{
    extern __shared__ float smem[];
    float* sm_h  = smem;                    // [BM][HS]
    float* sm_h1 = sm_h  + BM * HS;         // [BM][H1S]
    float* sm_h2 = sm_h1 + BM * H1S;        // [BM][H2S]

    const int tid   = threadIdx.x;
    const int wave  = tid >> 5;
    const int lane  = tid & 31;
    const int l16   = lane & 15;
    const int lhalf = lane >> 4;            // 0: lanes 0-15, 1: lanes 16-31

    const size_t rowBase = (size_t)blockIdx.x * BM;

    // ---------------- Phase 0: 3x3 valid conv into LDS ----------------
    const float k0 = cw[0], k1 = cw[1], k2 = cw[2];
    const float k3 = cw[3], k4 = cw[4], k5 = cw[5];
    const float k6 = cw[6], k7 = cw[7], k8 = cw[8];
    const float* xblk = x + rowBase * 784;

    // 64*676 = 43264 outputs, 256 threads -> exactly 169 per thread (no divergence)
    for (int o = tid; o < BM * 676; o += 256) {
        int r  = o / 676;
        int cc = o - r * 676;
        int oy = cc / 26;
        int ox = cc - oy * 26;
        const float* p = xblk + r * 784 + oy * 28 + ox;
        float s = p[0]  * k0 + p[1]  * k1 + p[2]  * k2
                + p[28] * k3 + p[29] * k4 + p[30] * k5
                + p[56] * k6 + p[57] * k7 + p[58] * k8;
        sm_h[r * HS + cc] = s;
    }
    __syncthreads();

    const v8f vzero = {0.f, 0.f, 0.f, 0.f, 0.f, 0.f, 0.f, 0.f};

    // ---------------- Phase 1: h1 = relu(h @ w1 + b1) via f32 WMMA ----------------
    // wave owns N columns [wave*32, wave*32+32): 2 N-tiles x 4 M-tiles
    {
        v8f acc[4][2];
        #pragma unroll
        for (int mt = 0; mt < 4; ++mt)
            #pragma unroll
            for (int nt = 0; nt < 2; ++nt)
                acc[mt][nt] = vzero;

        const int nB = wave * 32;
        for (int k = 0; k < 676; k += 4) {
            const int kk = k + 2 * lhalf;   // A/B lane-half K offset
            v2f bf[2];
            #pragma unroll
            for (int nt = 0; nt < 2; ++nt) {
                const int col = nB + nt * 16 + l16;
                bf[nt].x = w1[kk       * 256 + col];
                bf[nt].y = w1[(kk + 1) * 256 + col];
            }
            #pragma unroll
            for (int mt = 0; mt < 4; ++mt) {
                const int row = mt * 16 + l16;
                v2f af;
                af.x = sm_h[row * HS + kk];
                af.y = sm_h[row * HS + kk + 1];
                #pragma unroll
                for (int nt = 0; nt < 2; ++nt) {
                    acc[mt][nt] = __builtin_amdgcn_wmma_f32_16x16x4_f32(
                        false, af, false, bf[nt], (short)0, acc[mt][nt], false, false);
                }
            }
        }

        // store C fragments: VGPR j, lane<16 -> M=j, lane>=16 -> M=j+8
        #pragma unroll
        for (int nt = 0; nt < 2; ++nt) {
            const int col  = nB + nt * 16 + l16;
            const float bb = b1[col];
            #pragma unroll
            for (int mt = 0; mt < 4; ++mt) {
                #pragma unroll
                for (int j = 0; j < 8; ++j) {
                    const int row = mt * 16 + j + 8 * lhalf;
                    float v = acc[mt][nt][j] + bb;
                    sm_h1[row * H1S + col] = v > 0.f ? v : 0.f;
                }
            }
        }
    }
    __syncthreads();

    // ---------------- Phase 2: h2 = relu(h1 @ w2 + b2) via f32 WMMA ----------------
    // wave owns N columns [wave*16, wave*16+16): 1 N-tile x 4 M-tiles
    {
        v8f acc[4];
        #pragma unroll
        for (int mt = 0; mt < 4; ++mt) acc[mt] = vzero;

        const int col = wave * 16 + l16;
        for (int k = 0; k < 256; k += 4) {
            const int kk = k + 2 * lhalf;
            v2f bf;
            bf.x = w2[kk       * 128 + col];
            bf.y = w2[(kk + 1) * 128 + col];
            #pragma unroll
            for (int mt = 0; mt < 4; ++mt) {
                const int row = mt * 16 + l16;
                v2f af;
                af.x = sm_h1[row * H1S + kk];
                af.y = sm_h1[row * H1S + kk + 1];
                acc[mt] = __builtin_amdgcn_wmma_f32_16x16x4_f32(
                    false, af, false, bf, (short)0, acc[mt], false, false);
            }
        }

        const float bb = b2[col];
        #pragma unroll
        for (int mt = 0; mt < 4; ++mt) {
            #pragma unroll
            for (int j = 0; j < 8; ++j) {
                const int row = mt * 16 + j + 8 * lhalf;
                float v = acc[mt][j] + bb;
                sm_h2[row * H2S + col] = v > 0.f ? v : 0.f;
            }
        }
    }
    __syncthreads();

    // ---------------- Phase 3: out = h2 @ w3 + b3 (tiny, VALU) ----------------
    for (int o = tid; o < BM * 10; o += 256) {
        const int r = o / 10;
        const int c = o - r * 10;
        float s = b3[c];
        #pragma unroll 8
        for (int k = 0; k < 128; ++k)
            s = fmaf(sm_h2[r * H2S + k], w3[k * 10 + c], s);
        out[(rowBase + r) * 10 + c] = s;
    }
}

extern "C" void kernel_launch(void* const* d_in, const int* in_sizes, int n_in,
                              void* d_out, int out_size, void* d_ws, size_t ws_size,
                              hipStream_t stream) {
    const float* x  = (const float*)d_in[0];
    const float* cw = (const float*)d_in[1];
    const float* w1 = (const float*)d_in[2];
    const float* b1 = (const float*)d_in[3];
    const float* w2 = (const float*)d_in[4];
    const float* b2 = (const float*)d_in[5];
    const float* w3 = (const float*)d_in[6];
    const float* b3 = (const float*)d_in[7];
    float* out = (float*)d_out;

    const int B = in_sizes[0] / 784;          // 65536
    const size_t shmem = (size_t)LDS_FLOATS * sizeof(float);  // ~268 KB < 320 KB/WGP

    hipFuncSetAttribute((const void*)digitnet_fused_kernel,
                        hipFuncAttributeMaxDynamicSharedMemorySize, (int)shmem);

    dim3 grid(B / BM), block(256);
    digitnet_fused_kernel<<<grid, block, shmem, stream>>>(x, cw, w1, b1, w2, b2, w3, b3, out);
}